// AttentionWeight_14491219657298
// MI455X (gfx1250) — compile-verified
//
#include <hip/hip_runtime.h>
#include <hip/hip_bf16.h>

#define IN_F 256   // in_feats
#define NH   8     // heads
#define OF   64    // out_feats per head
#define EF   64    // edge_feats
#define NT   8     // edge types

typedef float v2f __attribute__((ext_vector_type(2)));
typedef float v8f __attribute__((ext_vector_type(8)));

// ---------------------------------------------------------------- utilities
__global__ void k_zero(float* __restrict__ p, int n) {
  int i = blockIdx.x * blockDim.x + threadIdx.x;
  if (i < n) p[i] = 0.0f;
}

// ---------------------------------------------------- fold W_fc with attn_l/r
// w_lr[k][col], col<8 -> Σ_o W_fc[k, col*64+o]*attn_l[col,o]
//               col>=8 -> same with attn_r, head = col-8
__global__ void k_prep_wlr(const float* __restrict__ W_fc,
                           const float* __restrict__ attn_l,
                           const float* __restrict__ attn_r,
                           float* __restrict__ w_lr) {
  int idx = blockIdx.x * blockDim.x + threadIdx.x;   // 256*16 = 4096 threads
  if (idx >= IN_F * 16) return;
  int k = idx >> 4, col = idx & 15;
  int h = col & 7;
  const float* av   = (col < 8) ? attn_l : attn_r;
  const float* wrow = W_fc + (size_t)k * (NH * OF) + h * OF;
  const float* arow = av + h * OF;
  float acc = 0.f;
  #pragma unroll 8
  for (int o = 0; o < OF; ++o) acc = fmaf(wrow[o], arow[o], acc);
  w_lr[k * 16 + col] = acc;
}

// -------------------------------- fold edge_emb @ W_e with attn_e -> [T,H]
__global__ void k_prep_ee(const float* __restrict__ edge_emb,
                          const float* __restrict__ W_e,
                          const float* __restrict__ attn_e,
                          float* __restrict__ ee) {
  int tid = threadIdx.x;           // 64 threads: (t,h)
  if (tid >= NT * NH) return;
  int t = tid >> 3, h = tid & 7;
  const float* arow = attn_e + h * EF;
  float acc = 0.f;
  for (int c = 0; c < EF; ++c) {
    const float* wrow = W_e + (size_t)c * (NH * EF) + h * EF;
    float inner = 0.f;
    #pragma unroll 8
    for (int f = 0; f < EF; ++f) inner = fmaf(wrow[f], arow[f], inner);
    acc = fmaf(edge_emb[t * EF + c], inner, acc);
  }
  ee[t * NH + h] = acc;
}

// --------------------------- WMMA GEMM: el_er[N][16] = feat[N][256] @ w_lr
// One wave per 16-row tile. V_WMMA_F32_16X16X4_F32, K loop of 64 steps.
// A 16x4 f32 layout: lane(row=l&15, half=l>>4) holds K=2*half, 2*half+1 (contiguous pair)
// B 4x16  f32 layout: mirrors A: B[k+2*half+{0,1}][col], col = l&15
// C/D 16x16 f32: VGPR j = out[M = j + 8*half][col]
__global__ __launch_bounds__(32) void k_gemm(const float* __restrict__ feat,
                                             const float* __restrict__ w_lr,
                                             float* __restrict__ el_er, int n) {
  const int lane = threadIdx.x;
  const int col  = lane & 15;
  const int half = lane >> 4;
  const int tile = blockIdx.x;

  int r = tile * 16 + col;                 // A row for this lane
  if (r >= n) r = n - 1;                   // clamp (N is a multiple of 16 anyway)
  const float* arow = feat + (size_t)r * IN_F;

  v8f c = {0.f, 0.f, 0.f, 0.f, 0.f, 0.f, 0.f, 0.f};
  #pragma unroll 4
  for (int k = 0; k < IN_F; k += 4) {
    v2f a = *(const v2f*)(arow + k + 2 * half);            // global_load_b64
    v2f b;
    b.x = w_lr[(k + 2 * half)     * 16 + col];
    b.y = w_lr[(k + 2 * half + 1) * 16 + col];
    c = __builtin_amdgcn_wmma_f32_16x16x4_f32(
        /*neg_a=*/false, a, /*neg_b=*/false, b,
        /*c_mod=*/(short)0, c, /*reuse_a=*/false, /*reuse_b=*/false);
  }

  // Epilogue: full tiles (the only case for N=100000) take the branch-free
  // path — 8 contiguous global_store_b32 per lane, 64B bursts per half-wave.
  if (tile * 16 + 15 < n) {
    float* orow = el_er + ((size_t)(tile * 16 + 8 * half)) * 16 + col;
    #pragma unroll
    for (int j = 0; j < 8; ++j) orow[(size_t)j * 16] = c[j];
  } else {
    #pragma unroll
    for (int j = 0; j < 8; ++j) {
      int orow = tile * 16 + j + 8 * half;
      if (orow < n) el_er[(size_t)orow * 16 + col] = c[j];
    }
  }
}

// ------------------- edge logits + ReLU + segment max (uint atomic, v >= 0)
__global__ void k_logits(const int* __restrict__ src, const int* __restrict__ dst,
                         const int* __restrict__ ety,
                         const float* __restrict__ el_er, const float* __restrict__ ee,
                         float* __restrict__ e_ws, float* __restrict__ mx, int E) {
  int tid = blockIdx.x * blockDim.x + threadIdx.x;
  if (tid >= E * NH) return;
  int e = tid >> 3, h = tid & 7;
  int s = src[e], d = dst[e], t = ety[e];
  float v = el_er[(size_t)s * 16 + h] + el_er[(size_t)d * 16 + 8 + h] + ee[t * NH + h];
  v = fmaxf(v, 0.f);                       // nn.ReLU in the reference
  e_ws[tid] = v;
  // v >= 0 and mx initialized to +0.0f: uint-bit max == float max
  atomicMax((unsigned int*)(mx + (size_t)d * NH + h), __float_as_uint(v));
}

// -------------------------------------- exp(e - m[dst]) + segment sum
__global__ void k_expsum(const int* __restrict__ dst,
                         const float* __restrict__ e_ws, const float* __restrict__ mx,
                         float* __restrict__ outv, float* __restrict__ ssum, int E) {
  int tid = blockIdx.x * blockDim.x + threadIdx.x;
  if (tid >= E * NH) return;
  int e = tid >> 3, h = tid & 7;
  int d = dst[e];
  float ex = __expf(e_ws[tid] - mx[(size_t)d * NH + h]);
  outv[tid] = ex;
  atomicAdd(ssum + (size_t)d * NH + h, ex);
}

// ------------------------------------------------ normalize: ex / s[dst]
__global__ void k_norm(const int* __restrict__ dst, const float* __restrict__ ssum,
                       float* __restrict__ outv, int E) {
  int tid = blockIdx.x * blockDim.x + threadIdx.x;
  if (tid >= E * NH) return;
  int e = tid >> 3, h = tid & 7;
  outv[tid] = outv[tid] / ssum[(size_t)dst[e] * NH + h];
}

extern "C" void kernel_launch(void* const* d_in, const int* in_sizes, int n_in,
                              void* d_out, int out_size, void* d_ws, size_t ws_size,
                              hipStream_t stream) {
  const float* feat     = (const float*)d_in[0];
  const int*   etype    = (const int*)d_in[1];
  const int*   src      = (const int*)d_in[2];
  const int*   dst      = (const int*)d_in[3];
  const float* W_fc     = (const float*)d_in[4];
  const float* edge_emb = (const float*)d_in[5];
  const float* W_e      = (const float*)d_in[6];
  const float* attn_l   = (const float*)d_in[7];
  const float* attn_r   = (const float*)d_in[8];
  const float* attn_e   = (const float*)d_in[9];
  float* outv = (float*)d_out;

  const int N = in_sizes[0] / IN_F;   // 100000
  const int E = in_sizes[1];          // 3200000

  // workspace layout (floats)
  float* ws    = (float*)d_ws;
  float* w_lr  = ws;                            // 256*16 = 4096
  float* ee    = ws + 4096;                     // 64 (padded)
  float* el_er = ws + 4160;                     // N*16
  float* mx    = el_er + (size_t)N * 16;        // N*8
  float* ssum  = mx + (size_t)N * NH;           // N*8
  float* e_ws  = ssum + (size_t)N * NH;         // E*8

  // 1) zero segment accumulators (mx, ssum are contiguous) — every call
  int zn = N * NH * 2;
  k_zero<<<dim3((zn + 255) / 256), dim3(256), 0, stream>>>(mx, zn);

  // 2) fold weight matrices
  k_prep_wlr<<<dim3((IN_F * 16 + 255) / 256), dim3(256), 0, stream>>>(
      W_fc, attn_l, attn_r, w_lr);
  k_prep_ee<<<dim3(1), dim3(64), 0, stream>>>(edge_emb, W_e, attn_e, ee);

  // 3) WMMA GEMM -> el_er[N][16]
  int ntiles = (N + 15) / 16;
  k_gemm<<<dim3(ntiles), dim3(32), 0, stream>>>(feat, w_lr, el_er, N);

  // 4-6) edge phase
  int tot = E * NH;
  dim3 g((tot + 255) / 256), b(256);
  k_logits<<<g, b, 0, stream>>>(src, dst, etype, el_er, ee, e_ws, mx, E);
  k_expsum<<<g, b, 0, stream>>>(dst, e_ws, mx, outv, ssum, E);
  k_norm<<<g, b, 0, stream>>>(dst, ssum, outv, E);
}